// MedicalMoESystem_68831145885978
// MI455X (gfx1250) — compile-verified
//
#include <hip/hip_runtime.h>
#include <math.h>

typedef float v2f __attribute__((ext_vector_type(2)));
typedef float v8f __attribute__((ext_vector_type(8)));

#define N_SAMPLES 1024
#define DIM       384
#define R_HID     128
#define NE        32
#define H1        256
#define H2        128
#define VOCAB     19756
#define NCOEF     8
#define K1AUG     (DIM + DIM * NCOEF)   // 3456
#define K2AUG     (H1 + H1 * NCOEF)     // 2304
#define MAX_TILES 96

// ---------------------------------------------------------------------------
// Router: logits = relu(x @ r_w1^T + b1) @ r_w2^T + b2 ; argmax (first max)
// ---------------------------------------------------------------------------
__global__ __launch_bounds__(R_HID)
void router_kernel(const float* __restrict__ x,
                   const float* __restrict__ w1, const float* __restrict__ b1,
                   const float* __restrict__ w2, const float* __restrict__ b2,
                   int* __restrict__ eidx) {
  __shared__ float xs[DIM];
  __shared__ float hs[R_HID];
  __shared__ float lg[NE];
  const int i = blockIdx.x;
  for (int d = threadIdx.x; d < DIM; d += blockDim.x) xs[d] = x[i * DIM + d];
  __syncthreads();
  const int j = threadIdx.x;
  float acc = b1[j];
  for (int d = 0; d < DIM; ++d) acc = fmaf(xs[d], w1[j * DIM + d], acc);
  hs[j] = fmaxf(acc, 0.0f);
  __syncthreads();
  if (j < NE) {
    float a2 = b2[j];
    for (int k = 0; k < R_HID; ++k) a2 = fmaf(hs[k], w2[j * R_HID + k], a2);
    lg[j] = a2;
  }
  __syncthreads();
  if (j == 0) {
    int best = 0; float bv = lg[0];
    for (int e = 1; e < NE; ++e) if (lg[e] > bv) { bv = lg[e]; best = e; }
    eidx[i] = best;
  }
}

// ---------------------------------------------------------------------------
// Stable group-by-expert + 16-row tile table (single thread; trivial cost)
// ---------------------------------------------------------------------------
__global__ void group_build_kernel(const int* __restrict__ eidx,
                                   int* __restrict__ order,
                                   int* __restrict__ off,
                                   int* __restrict__ tile_e,
                                   int* __restrict__ tile_rs,
                                   int* __restrict__ tile_cnt) {
  if (threadIdx.x != 0 || blockIdx.x != 0) return;
  int cnt[NE];
  for (int e = 0; e < NE; ++e) cnt[e] = 0;
  for (int i = 0; i < N_SAMPLES; ++i) cnt[eidx[i]]++;
  off[0] = 0;
  for (int e = 0; e < NE; ++e) off[e + 1] = off[e] + cnt[e];
  int cur[NE];
  for (int e = 0; e < NE; ++e) cur[e] = off[e];
  for (int i = 0; i < N_SAMPLES; ++i) order[cur[eidx[i]]++] = i;  // stable
  int t = 0;
  for (int e = 0; e < NE; ++e)
    for (int r = off[e]; r < off[e + 1]; r += 16) { tile_e[t] = e; tile_rs[t] = r; ++t; }
  *tile_cnt = t;   // <= 96
}

// ---------------------------------------------------------------------------
// Cubic B-spline bases (efficient-KAN): uniform grid on [-1,1], 5 cells,
// order 3 -> 8 coefficients. Matches Cox-de Boor recursion in the reference.
// ---------------------------------------------------------------------------
__device__ inline void bsplines8(float xv, float* __restrict__ out) {
  const float h = 0.4f;
  float t[12];
#pragma unroll
  for (int j = 0; j < 12; ++j) t[j] = -1.0f + (float)(j - 3) * h;
  float b[11];
#pragma unroll
  for (int j = 0; j < 11; ++j) b[j] = (xv >= t[j] && xv < t[j + 1]) ? 1.0f : 0.0f;
#pragma unroll
  for (int d = 1; d <= 3; ++d) {
#pragma unroll 10
    for (int j = 0; j < 11 - d; ++j) {
      float left  = (xv - t[j]) / (t[j + d] - t[j]) * b[j];
      float right = (t[j + d + 1] - xv) / (t[j + d + 1] - t[j + 1]) * b[j + 1];
      b[j] = left + right;
    }
  }
#pragma unroll
  for (int c = 0; c < NCOEF; ++c) out[c] = b[c];
}

// Augmented activation row: [ silu(x) (indim) | bases (indim*8) ]
__global__ void prep_kernel(const float* __restrict__ in, float* __restrict__ xa,
                            int indim, int kaug) {
  const int i = blockIdx.x;
  const int d = threadIdx.x;      // blockDim.x == indim
  const float xv = in[(size_t)i * indim + d];
  const float s  = xv / (1.0f + expf(-xv));
  float* row = xa + (size_t)i * kaug;
  row[d] = s;
  float bb[NCOEF];
  bsplines8(xv, bb);
#pragma unroll
  for (int c = 0; c < NCOEF; ++c) row[indim + d * NCOEF + c] = bb[c];
}

// ---------------------------------------------------------------------------
// Grouped-GEMM tile kernel: one wave computes a 16(M rows) x 16(N cols) tile
// with V_WMMA_F32_16X16X4_F32 over K = K0 (+ K1). Rows gathered via order[],
// B from per-expert weights W0 ([E,COLS,K0]) then W1 ([E,COLS,K1]).
// WMMA f32 operand layout (ISA 7.12.2):
//   A  : lane (l&15) = M row; VGPR0/1 = K = k0+2*(l>>4) (+1)
//   B  : lane (l&15) = N col; VGPR0/1 = K = k0+2*(l>>4) (+1)
//   C/D: VGPR i -> M = i + 8*(l>>4), N = l&15
// ---------------------------------------------------------------------------
__global__ __launch_bounds__(32)
void moe_wmma_gemm(const float* __restrict__ A,     // [N, K0+K1] by original row
                   const float* __restrict__ W0,    // [E, COLS, K0]
                   const float* __restrict__ W1,    // [E, COLS, K1] or nullptr
                   const float* __restrict__ bias,  // [E, COLS] or nullptr
                   const int* __restrict__ order,
                   const int* __restrict__ off,
                   const int* __restrict__ tile_e,
                   const int* __restrict__ tile_rs,
                   const int* __restrict__ tile_cnt,
                   float* __restrict__ Out,         // [N, COLS] by original row
                   int K0, int K1, int COLS) {
  const int slot = blockIdx.y;
  if (slot >= *tile_cnt) return;                 // wave-uniform early exit
  const int e    = tile_e[slot];
  const int rs   = tile_rs[slot];
  const int rend = off[e + 1];

  const int lane = threadIdx.x;                  // 0..31
  const int half = lane >> 4;                    // 0/1 -> K sub-pair
  const int lm   = lane & 15;                    // M row / N col within tile

  // Gathered A row (invalid rows duplicate the first row of the segment)
  const int  ridx  = rs + lm;
  const bool rvld  = ridx < rend;
  const int  srow  = order[rvld ? ridx : rs];
  const int  Ktot  = K0 + K1;
  const float* arow = A + (size_t)srow * Ktot;

  const int col  = blockIdx.x * 16 + lm;
  const int ccol = (col < COLS) ? col : (COLS - 1);
  const float* w0col = W0 + ((size_t)e * COLS + ccol) * K0;
  const float* w1col = W1 ? (W1 + ((size_t)e * COLS + ccol) * K1) : nullptr;

  v8f acc = {};
#pragma unroll 4
  for (int k0 = 0; k0 < Ktot; k0 += 4) {
    const int ka = k0 + 2 * half;
    v2f a, b;
    a.x = arow[ka];
    a.y = arow[ka + 1];
    if (k0 < K0) {                               // K0 % 4 == 0 -> wave-uniform
      b.x = w0col[ka];
      b.y = w0col[ka + 1];
    } else {
      const int kk = ka - K0;
      b.x = w1col[kk];
      b.y = w1col[kk + 1];
    }
    acc = __builtin_amdgcn_wmma_f32_16x16x4_f32(
        /*neg_a=*/false, a, /*neg_b=*/false, b,
        /*c_mod=*/(short)0, acc, /*reuse_a=*/false, /*reuse_b=*/false);
  }

  const float bval  = bias ? bias[(size_t)e * COLS + ccol] : 0.0f;
  const bool  cvld  = col < COLS;
#pragma unroll
  for (int v = 0; v < 8; ++v) {
    const int m  = v + 8 * half;                 // D layout
    const int rr = rs + m;
    if (cvld && rr < rend) {
      const int so = order[rr];
      Out[(size_t)so * COLS + col] = acc[v] + bval;
    }
  }
}

// ---------------------------------------------------------------------------
// Host-side launch
// ---------------------------------------------------------------------------
extern "C" void kernel_launch(void* const* d_in, const int* in_sizes, int n_in,
                              void* d_out, int out_size, void* d_ws, size_t ws_size,
                              hipStream_t stream) {
  const float* x         = (const float*)d_in[0];
  const float* r_w1      = (const float*)d_in[1];
  const float* r_b1      = (const float*)d_in[2];
  const float* r_w2      = (const float*)d_in[3];
  const float* r_b2      = (const float*)d_in[4];
  const float* base_w1   = (const float*)d_in[5];   // [E,256,384]
  const float* spline_w1 = (const float*)d_in[6];   // [E,256,384,8]
  const float* base_w2   = (const float*)d_in[7];   // [E,128,256]
  const float* spline_w2 = (const float*)d_in[8];   // [E,128,256,8]
  const float* final_w   = (const float*)d_in[9];   // [E,19756,128]
  const float* final_b   = (const float*)d_in[10];  // [E,19756]
  float* out = (float*)d_out;

  char* ws = (char*)d_ws;
  size_t o = 0;
  int* eidx     = (int*)(ws + o); o += N_SAMPLES * sizeof(int);
  int* order    = (int*)(ws + o); o += N_SAMPLES * sizeof(int);
  int* off      = (int*)(ws + o); o += 64 * sizeof(int);
  int* tile_e   = (int*)(ws + o); o += 128 * sizeof(int);
  int* tile_rs  = (int*)(ws + o); o += 128 * sizeof(int);
  int* tile_cnt = (int*)(ws + o); o += 128 * sizeof(int);
  o = (o + 255) & ~(size_t)255;
  float* xa1 = (float*)(ws + o); o += (size_t)N_SAMPLES * K1AUG * sizeof(float);
  float* f1  = (float*)(ws + o); o += (size_t)N_SAMPLES * H1    * sizeof(float);
  float* xa2 = (float*)(ws + o); o += (size_t)N_SAMPLES * K2AUG * sizeof(float);
  float* f2  = (float*)(ws + o); o += (size_t)N_SAMPLES * H2    * sizeof(float);
  (void)ws_size; (void)in_sizes; (void)n_in; (void)out_size;

  // 1) router
  router_kernel<<<N_SAMPLES, R_HID, 0, stream>>>(x, r_w1, r_b1, r_w2, r_b2, eidx);
  // 2) group by expert + tile table
  group_build_kernel<<<1, 1, 0, stream>>>(eidx, order, off, tile_e, tile_rs, tile_cnt);
  // 3) KAN layer 1: augment then grouped GEMM (K = 384 + 3072)
  prep_kernel<<<N_SAMPLES, DIM, 0, stream>>>(x, xa1, DIM, K1AUG);
  {
    dim3 grid(H1 / 16, MAX_TILES);
    moe_wmma_gemm<<<grid, 32, 0, stream>>>(xa1, base_w1, spline_w1, nullptr,
                                           order, off, tile_e, tile_rs, tile_cnt,
                                           f1, DIM, DIM * NCOEF, H1);
  }
  // 4) KAN layer 2 (K = 256 + 2048)
  prep_kernel<<<N_SAMPLES, H1, 0, stream>>>(f1, xa2, H1, K2AUG);
  {
    dim3 grid(H2 / 16, MAX_TILES);
    moe_wmma_gemm<<<grid, 32, 0, stream>>>(xa2, base_w2, spline_w2, nullptr,
                                           order, off, tile_e, tile_rs, tile_cnt,
                                           f2, H1, H1 * NCOEF, H2);
  }
  // 5) final vocab projection (dominant, HBM-bound: 323 MB weights + 81 MB out)
  {
    dim3 grid((VOCAB + 15) / 16, MAX_TILES);
    moe_wmma_gemm<<<grid, 32, 0, stream>>>(f2, final_w, nullptr, final_b,
                                           order, off, tile_e, tile_rs, tile_cnt,
                                           out, H2, 0, VOCAB);
  }
}